// Weight_Vec_Ent_Sink_70050916598304
// MI455X (gfx1250) — compile-verified
//
#include <hip/hip_runtime.h>
#include <math.h>

typedef __attribute__((ext_vector_type(2))) float v2f;
typedef __attribute__((ext_vector_type(8))) float v8f;

#define MM 4096
#define NN 1024
#define SIGMAF 0.1f
#define EPSF 1e-7f
#define TOLF 1e-4f

// ---------------- init: p=f=0 (M), g=0 (N), done=0 ----------------
__global__ __launch_bounds__(256) void k_init(float* p, float* f, float* g, int* done) {
  int i = blockIdx.x * 256 + threadIdx.x;
  if (i < MM) { p[i] = 0.0f; f[i] = 0.0f; }
  if (i < NN) g[i] = 0.0f;
  if (i == 0) *done = 0;
}

// ---------------- Gibbs kernel: K = exp(-D/sigma) ----------------
__global__ __launch_bounds__(256) void k_expneg(const float* __restrict__ D,
                                                float* __restrict__ K, int n) {
  int i = blockIdx.x * 256 + threadIdx.x;
  if (i < n) K[i] = expf(-10.0f * D[i]);
}

// ---------------- KcT[m][n] = exp(-10 * cross[n][m]) (tiled transpose) -------
__global__ __launch_bounds__(256) void k_exp_transpose(const float* __restrict__ cross,
                                                       float* __restrict__ KcT) {
  __shared__ float tile[32][33];
  int m0 = blockIdx.x << 5;
  int n0 = blockIdx.y << 5;
  int tx = threadIdx.x, ty = threadIdx.y;
#pragma unroll
  for (int j = 0; j < 4; ++j) {
    int a = ty + j * 8;
    tile[a][tx] = expf(-10.0f * cross[(size_t)(n0 + a) * MM + m0 + tx]);
  }
  __syncthreads();
#pragma unroll
  for (int j = 0; j < 4; ++j) {
    int a = ty + j * 8;
    KcT[(size_t)(m0 + a) * NN + n0 + tx] = tile[tx][a];
  }
}

// ---------------- out[i] = w[i] * exp(x[i]) ----------------
__global__ __launch_bounds__(256) void k_wexp(const float* __restrict__ w,
                                              const float* __restrict__ x,
                                              float* __restrict__ out, int n) {
  int i = blockIdx.x * 256 + threadIdx.x;
  if (i < n) out[i] = w[i] * expf(x[i]);
}

// ---------------- GEMV via V_WMMA_F32_16X16X4_F32 ----------------
// s[c] = sum_r Kmat[r][c] * v[r]   (column reduction of row-major [R,C])
// A (16x4) = broadcast of v[r..r+3] into all 16 rows; B (4x16) = K tile.
// A vgpr j: lanes 0-15 hold k=j, lanes 16-31 k=j+2 (ISA layout);
// B assumed mirrored (lane = column, vgpr j: k=j / k=j+2 for lane halves).
// v is staged global->LDS with async-to-LDS DMA (ASYNCcnt).
// Epilogue modes: 0: pnew=0.5*(p+ct), |diff| partials
//                 1: out=ct*sigma
//                 2: out=ct/(1+sigma)
//                 3: out=ct, |diff vs cmp| partials
__global__ __launch_bounds__(256) void k_gemv_wmma(const float* __restrict__ Kmat,
                                                   int R, int C,
                                                   const float* __restrict__ vvec,
                                                   float* __restrict__ outv,
                                                   const float* __restrict__ cmpv,
                                                   float* __restrict__ partials,
                                                   int mode) {
  __shared__ float lds_v[MM];
  __shared__ float lds_part[8 * 16];
  int tid = threadIdx.x;
  int wave = tid >> 5;
  int lane = tid & 31;
  int c0 = blockIdx.x << 4;

  // ---- async global->LDS fill of v (R is a multiple of 1024; 16B aligned) ----
  {
    __attribute__((address_space(3))) float* as3 =
        (__attribute__((address_space(3))) float*)lds_v;
    int chunks = R >> 10;                      // R/1024 (1024 floats per pass)
    for (int ch = 0; ch < chunks; ++ch) {
      unsigned elem = ((unsigned)ch << 10) + ((unsigned)tid << 2);
      unsigned ldsOff = (unsigned)(uintptr_t)(as3 + elem);   // byte offset in LDS
      unsigned gOff = elem * 4u;                             // byte offset in global
      asm volatile("global_load_async_to_lds_b128 %0, %1, %2"
                   :
                   : "v"(ldsOff), "v"(gOff), "s"(vvec)
                   : "memory");
    }
    asm volatile("s_wait_asynccnt 0x0" ::: "memory");
  }
  __syncthreads();

  int rowsPer = R >> 3;            // 8 waves split the rows (multiple of 16)
  int rbeg = wave * rowsPer;
  int hi2 = (lane >> 4) << 1;      // 0 for lanes 0-15, 2 for lanes 16-31
  const float* base = Kmat + (size_t)c0 + (lane & 15);

  // 4 independent accumulators break the WMMA D->C dependence chain
  v8f accs[4] = {{}, {}, {}, {}};
  for (int r = rbeg; r < rbeg + rowsPer; r += 16) {
#pragma unroll
    for (int u = 0; u < 4; ++u) {
      int r0 = r + 4 * u + hi2;
      v2f a, b;
      a[0] = lds_v[r0];
      a[1] = lds_v[r0 + 1];
      b[0] = base[(size_t)r0 * C];
      b[1] = base[(size_t)(r0 + 1) * C];
      accs[u] = __builtin_amdgcn_wmma_f32_16x16x4_f32(false, a, false, b,
                                                      (short)0, accs[u],
                                                      false, false);
    }
  }

  // all 16 D rows identical -> lane j (j<16) holds partial s[c0+j] in acc[0]
  float sl = (accs[0][0] + accs[1][0]) + (accs[2][0] + accs[3][0]);
  if (lane < 16) lds_part[wave * 16 + lane] = sl;
  __syncthreads();

  if (tid < 16) {
    float s = 0.0f;
#pragma unroll
    for (int w = 0; w < 8; ++w) s += lds_part[w * 16 + tid];
    float ct = -logf(s + EPSF);
    int c = c0 + tid;
    float res, diff = 0.0f;
    if (mode == 0)      { res = 0.5f * (cmpv[c] + ct); diff = fabsf(res - cmpv[c]); }
    else if (mode == 1) { res = ct * SIGMAF; }
    else if (mode == 2) { res = ct / (1.0f + SIGMAF); }
    else                { res = ct; diff = fabsf(res - cmpv[c]); }
    outv[c] = res;
    lds_part[tid] = diff;   // same-wave lockstep: reads above precede this store
  }
  __syncthreads();
  if (tid == 0 && partials != nullptr) {
    float d = 0.0f;
#pragma unroll
    for (int i = 0; i < 16; ++i) d += lds_part[i];
    partials[blockIdx.x] = d;
  }
}

// ---------------- commit: reduce partials, update done, gated copies --------
__global__ __launch_bounds__(256) void k_commit(const float* __restrict__ partials,
                                                int nPart, float invMean,
                                                int* __restrict__ done,
                                                float* __restrict__ dst1,
                                                const float* __restrict__ src1, int n1,
                                                float* __restrict__ dst2,
                                                const float* __restrict__ src2, int n2) {
  __shared__ float red[256];
  __shared__ int dOld;
  int tid = threadIdx.x;
  float s = 0.0f;
  for (int i = tid; i < nPart; i += 256) s += partials[i];
  red[tid] = s;
  __syncthreads();
  for (int off = 128; off > 0; off >>= 1) {
    if (tid < off) red[tid] += red[tid + off];
    __syncthreads();
  }
  if (tid == 0) {
    dOld = *done;
    float cond = SIGMAF * red[0] * invMean;
    if (dOld == 0 && cond < TOLF) *done = 1;   // done |= cond
  }
  __syncthreads();
  if (!dOld) {
    for (int i = tid; i < n1; i += 256) dst1[i] = src1[i];
    if (dst2 != nullptr)
      for (int i = tid; i < n2; i += 256) dst2[i] = src2[i];
  }
}

// ---------------- final weighted sums ----------------
__global__ __launch_bounds__(256) void k_final(const float* __restrict__ f,
                                               const float* __restrict__ pa,
                                               const float* __restrict__ alpha,
                                               const float* __restrict__ g,
                                               const float* __restrict__ tm,
                                               float* __restrict__ out) {
  __shared__ float red[256];
  int tid = threadIdx.x;
  float s = 0.0f;
  for (int i = tid; i < MM; i += 256)
    s += ((1.0f - expf(-f[i] * SIGMAF)) - pa[i]) * alpha[i];
  for (int i = tid; i < NN; i += 256)
    s += tm[i] * (g[i] * SIGMAF);
  red[tid] = s;
  __syncthreads();
  for (int off = 128; off > 0; off >>= 1) {
    if (tid < off) red[tid] += red[tid + off];
    __syncthreads();
  }
  if (tid == 0) out[0] = red[0] * 1000.0f;
}

extern "C" void kernel_launch(void* const* d_in, const int* in_sizes, int n_in,
                              void* d_out, int out_size, void* d_ws, size_t ws_size,
                              hipStream_t stream) {
  const float* alpha = (const float*)d_in[0];   // pre_density [4096]
  const float* tmass = (const float*)d_in[1];   // target_mass [1024]
  const float* cross = (const float*)d_in[2];   // dis_mtx_cross [1024,4096]
  const float* dens  = (const float*)d_in[3];   // dis_mtx_density [4096,4096]
  float* out = (float*)d_out;

  float* ws   = (float*)d_ws;
  float* Kd   = ws;                              // [M*M]
  float* Kc   = Kd  + (size_t)MM * MM;           // [N*M]
  float* KcT  = Kc  + (size_t)NN * MM;           // [M*N]
  float* p    = KcT + (size_t)MM * NN;           // [M]
  float* pnew = p    + MM;
  float* f    = pnew + MM;
  float* fnew = f    + MM;
  float* g    = fnew + MM;                       // [N]
  float* gnew = g    + NN;
  float* vbuf = gnew + NN;                       // [M] scratch vector
  float* pa   = vbuf + MM;                       // [M] final p_auto
  float* parts = pa + MM;                        // [256] block partials
  int*   done  = (int*)(parts + 256);

  // init + precompute Gibbs kernels (once; everything then lives in L2)
  k_init<<<MM / 256, 256, 0, stream>>>(p, f, g, done);
  k_expneg<<<(MM * MM) / 256, 256, 0, stream>>>(dens, Kd, MM * MM);
  k_expneg<<<(NN * MM) / 256, 256, 0, stream>>>(cross, Kc, NN * MM);
  k_exp_transpose<<<dim3(MM / 32, NN / 32), dim3(32, 8), 0, stream>>>(cross, KcT);

  // self-OT: 20 iterations
  for (int it = 0; it < 20; ++it) {
    k_wexp<<<MM / 256, 256, 0, stream>>>(alpha, p, vbuf, MM);
    k_gemv_wmma<<<MM / 16, 256, 0, stream>>>(Kd, MM, MM, vbuf, pnew, p, parts, 0);
    k_commit<<<1, 256, 0, stream>>>(parts, MM / 16, 1.0f / MM, done,
                                    p, pnew, MM, nullptr, nullptr, 0);
  }

  // cross-OT Sinkhorn: 200 iterations
  for (int it = 0; it < 200; ++it) {
    k_wexp<<<NN / 256, 256, 0, stream>>>(tmass, g, vbuf, NN);
    k_gemv_wmma<<<MM / 16, 256, 0, stream>>>(Kc, NN, MM, vbuf, fnew, nullptr, nullptr, 2);
    k_wexp<<<MM / 256, 256, 0, stream>>>(alpha, fnew, vbuf, MM);
    k_gemv_wmma<<<NN / 16, 256, 0, stream>>>(KcT, MM, NN, vbuf, gnew, g, parts, 3);
    k_commit<<<1, 256, 0, stream>>>(parts, NN / 16, 1.0f / NN, done,
                                    f, fnew, MM, g, gnew, NN);
  }

  // final potentials + scalar
  k_wexp<<<MM / 256, 256, 0, stream>>>(alpha, p, vbuf, MM);
  k_gemv_wmma<<<MM / 16, 256, 0, stream>>>(Kd, MM, MM, vbuf, pa, nullptr, nullptr, 1);
  k_final<<<1, 256, 0, stream>>>(f, pa, alpha, g, tmass, out);
}